// HopfieldHNL_90185723281719
// MI455X (gfx1250) — compile-verified
//
#include <hip/hip_runtime.h>

typedef __attribute__((ext_vector_type(2))) float v2f;
typedef __attribute__((ext_vector_type(8))) float v8f;
typedef int v4i_vs __attribute__((vector_size(16)));   // matches builtin's V4i

#define B_    1024
#define INF_  512
#define H_    8
#define HD_   64
#define BIN_  2048
#define M_    16384
#define ZTOP  64

// k3 tiling
#define KC_   16              // K chunk (bins per stage)
#define SD_   20              // padded LDS row stride in floats (16 + 4)

// ---------------- CDNA5 async global->LDS path (guarded) ----------------
#if defined(__gfx1250__) && __has_builtin(__builtin_amdgcn_global_load_async_to_lds_b128)
#define HAVE_ASYNC_LDS 1
#define ASYNC_CP_B128(gp, lp)                                              \
  __builtin_amdgcn_global_load_async_to_lds_b128(                          \
      (__attribute__((address_space(1))) v4i_vs*)(gp),                     \
      (__attribute__((address_space(3))) v4i_vs*)(lp), 0, 0)
#if __has_builtin(__builtin_amdgcn_s_wait_asynccnt)
#define WAIT_ASYNC(n) __builtin_amdgcn_s_wait_asynccnt(n)
#else
#define WAIT_ASYNC(n) asm volatile("s_wait_asynccnt %0" ::"n"(n) : "memory")
#endif
#else
#define HAVE_ASYNC_LDS 0
#endif

// ---------------- helpers ----------------
__device__ __forceinline__ unsigned int f32_key(float f) {
  unsigned int u = __float_as_uint(f);
  return (u & 0x80000000u) ? ~u : (u | 0x80000000u);
}
__device__ __forceinline__ unsigned long long shflx64(unsigned long long v, int m) {
  unsigned int lo = (unsigned int)v;
  unsigned int hi = (unsigned int)(v >> 32);
  lo = __shfl_xor(lo, m, 32);
  hi = __shfl_xor(hi, m, 32);
  return ((unsigned long long)hi << 32) | lo;
}

// ---------------- kernel 1: Q = Wq@x + bq, per-head L2 normalize ----------------
__global__ void k1_qnorm(const float* __restrict__ x, const float* __restrict__ Wq,
                         const float* __restrict__ bq, float* __restrict__ qn) {
  const int bh = blockIdx.x;            // b*H + h
  const int b  = bh >> 3;
  const int h  = bh & 7;
  const int d  = threadIdx.x;           // 0..63
  const int row = h * HD_ + d;
  const float* wr = Wq + (size_t)row * INF_;
  const float* xr = x  + (size_t)b   * INF_;
  float acc = bq[row];
  for (int k = 0; k < INF_; k += 4) {
    acc = fmaf(wr[k+0], xr[k+0], acc);
    acc = fmaf(wr[k+1], xr[k+1], acc);
    acc = fmaf(wr[k+2], xr[k+2], acc);
    acc = fmaf(wr[k+3], xr[k+3], acc);
  }
  __shared__ float sq[HD_];
  sq[d] = acc * acc;
  __syncthreads();
  float s = 0.f;
  #pragma unroll
  for (int i = 0; i < HD_; ++i) s += sq[i];
  qn[(size_t)bh * HD_ + d] = acc / sqrtf(s);
}

// ---------------- kernel 2: bin scores + exact top-64 -> 0/1 mask ----------------
// grid (H_, B_/4), block 256.  keys LDS: 4*2048*4 = 32KB.
__global__ void __launch_bounds__(256) k2_topk(const float* __restrict__ bin_proj,
                                               const float* __restrict__ qn,
                                               float* __restrict__ mask) {
  const int h  = blockIdx.x;
  const int bg = blockIdx.y;            // group of 4 samples
  const int t  = threadIdx.x;
  const int lane = t & 31, wv = t >> 5;

  __shared__ float        sqn[4][HD_];
  __shared__ unsigned int keys[4][BIN_];

  for (int i = t; i < 4 * HD_; i += 256) {
    int bb = i >> 6, d = i & 63;
    sqn[bb][d] = qn[((size_t)(bg * 4 + bb) * H_ + h) * HD_ + d];
  }
  __syncthreads();

  const float* bp = bin_proj + (size_t)h * BIN_ * HD_;
  for (int it = 0; it < BIN_ / 256; ++it) {
    const int bin = it * 256 + t;
    const float* rowp = bp + (size_t)bin * HD_;
    float acc[4] = {0.f, 0.f, 0.f, 0.f};
    for (int c = 0; c < HD_; c += 4) {
      float4 pv = *(const float4*)(rowp + c);
      #pragma unroll
      for (int bb = 0; bb < 4; ++bb) {
        acc[bb] = fmaf(pv.x, sqn[bb][c+0], acc[bb]);
        acc[bb] = fmaf(pv.y, sqn[bb][c+1], acc[bb]);
        acc[bb] = fmaf(pv.z, sqn[bb][c+2], acc[bb]);
        acc[bb] = fmaf(pv.w, sqn[bb][c+3], acc[bb]);
      }
    }
    #pragma unroll
    for (int bb = 0; bb < 4; ++bb) keys[bb][bin] = f32_key(acc[bb]);
  }
  __syncthreads();

  if (wv < 4) {                          // wave wv selects for sample bb = wv
    const int bb = wv;
    // bitwise radix search for the 64th-largest key T (monotone -> greedy exact)
    unsigned int T = 0u;
    for (int bit = 31; bit >= 0; --bit) {
      const unsigned int cand = T | (1u << bit);
      int cnt = 0;
      for (int j = lane; j < BIN_; j += 32) cnt += (keys[bb][j] >= cand) ? 1 : 0;
      #pragma unroll
      for (int off = 16; off > 0; off >>= 1) cnt += __shfl_xor(cnt, off, 32);
      if (cnt >= ZTOP) T = cand;
    }
    int cgt = 0;
    for (int j = lane; j < BIN_; j += 32) cgt += (keys[bb][j] > T) ? 1 : 0;
    #pragma unroll
    for (int off = 16; off > 0; off >>= 1) cgt += __shfl_xor(cgt, off, 32);
    const int need_eq = ZTOP - cgt;      // >=1, ties filled lowest-index-first

    float* mrow = mask + ((size_t)h * B_ + (size_t)(bg * 4 + bb)) * BIN_;
    int taken = 0;
    for (int c = 0; c < BIN_ / 32; ++c) {
      const int j = c * 32 + lane;
      const unsigned int k = keys[bb][j];
      const bool gt = (k > T);
      const bool eq = (k == T);
      const unsigned int beq = (unsigned int)__ballot(eq);
      const int below = __popc(beq & ((1u << lane) - 1u));
      const bool sel = gt || (eq && (taken + below) < need_eq);
      mrow[j] = sel ? 1.0f : 0.0f;
      taken += __popc(beq);
    }
  }
}

// ---------------- kernel 3: attn = mask @ W^T via f32 WMMA, fused argmax ----------------
// grid (B_/128, M_/128, H_), block 256 (8 waves). C tile 128b x 128m.
// Double-buffered LDS staging (async global->LDS on CDNA5), K chunk 16,
// padded stride 20 floats (conflict-free v2f fragment reads). LDS = 4*10KB.
__global__ void __launch_bounds__(256) k3_attn(const float* __restrict__ Wm,
                                               const float* __restrict__ mask,
                                               unsigned long long* __restrict__ am) {
  const int h     = blockIdx.z;
  const int mbase = blockIdx.y * 128;
  const int bbase = blockIdx.x * 128;
  const int t     = threadIdx.x;
  const int lane  = t & 31, wv = t >> 5;
  const int Lr    = lane & 15, half = lane >> 4;

  __shared__ float lW[2][128 * SD_];
  __shared__ float lA[2][128 * SD_];

  const v8f zero = {0.f,0.f,0.f,0.f,0.f,0.f,0.f,0.f};
  v8f acc[8];
  #pragma unroll
  for (int nt = 0; nt < 8; ++nt) acc[nt] = zero;

  const float* Wh = Wm   + (size_t)h * M_ * BIN_;
  const float* Mh = mask + (size_t)h * B_ * BIN_;

  // per-thread staging assignment: 2 float4 per tile (512 float4 per 128x16 tile)
  const int e0 = t, e1 = t + 256;
  const int r0e = e0 >> 2, q0 = (e0 & 3) * 4;   // row, float offset
  const int r1e = e1 >> 2, q1 = (e1 & 3) * 4;

#if HAVE_ASYNC_LDS
  #define CP128(g, l) ASYNC_CP_B128(g, l)
#else
  #define CP128(g, l) (*(float4*)(l) = *(const float4*)(g))
#endif

  #define ISSUE_CHUNK(c, buf)                                                        \
    do {                                                                             \
      const int k0_ = (c) * KC_;                                                     \
      const float* gw0 = Wh + (size_t)(mbase + r0e) * BIN_ + k0_ + q0;               \
      const float* gw1 = Wh + (size_t)(mbase + r1e) * BIN_ + k0_ + q1;               \
      const float* ga0 = Mh + (size_t)(bbase + r0e) * BIN_ + k0_ + q0;               \
      const float* ga1 = Mh + (size_t)(bbase + r1e) * BIN_ + k0_ + q1;               \
      float* lw0 = &lW[buf][r0e * SD_ + q0];                                         \
      float* lw1 = &lW[buf][r1e * SD_ + q1];                                         \
      float* la0 = &lA[buf][r0e * SD_ + q0];                                         \
      float* la1 = &lA[buf][r1e * SD_ + q1];                                         \
      CP128(gw0, lw0); CP128(gw1, lw1); CP128(ga0, la0); CP128(ga1, la1);            \
    } while (0)

  ISSUE_CHUNK(0, 0);

  for (int c = 0; c < BIN_ / KC_; ++c) {
    const int p = c & 1;
    if (c + 1 < BIN_ / KC_) {
      ISSUE_CHUNK(c + 1, 1 - p);
#if HAVE_ASYNC_LDS
      WAIT_ASYNC(4);        // 4 newer ops in flight; the 4 for chunk c are done
#endif
    } else {
#if HAVE_ASYNC_LDS
      WAIT_ASYNC(0);
#endif
    }
    __syncthreads();        // chunk c fully staged in LDS for all waves

    #pragma unroll
    for (int k1 = 0; k1 < KC_; k1 += 4) {
      // A 16x4 f32: lanes 0-15 hold K={k1,k1+1}, lanes 16-31 hold K={k1+2,k1+3}
      v2f a = *(const v2f*)&lA[p][(wv * 16 + Lr) * SD_ + k1 + 2 * half];
      #pragma unroll
      for (int nt = 0; nt < 8; ++nt) {
        v2f bf = *(const v2f*)&lW[p][(nt * 16 + Lr) * SD_ + k1 + 2 * half];
        acc[nt] = __builtin_amdgcn_wmma_f32_16x16x4_f32(
            false, a, false, bf, (short)0, acc[nt], false, false);
      }
    }
    __syncthreads();        // all waves done reading buf p; safe to overwrite
  }
  #undef CP128
  #undef ISSUE_CHUNK

  // Epilogue: per lane, C/D VGPR r holds attn[b=bbase+wv*16+8*half+r][m=mbase+nt*16+Lr].
  #pragma unroll
  for (int r = 0; r < 8; ++r) {
    float v  = acc[0][r];
    int   bm = mbase + Lr;
    #pragma unroll
    for (int nt = 1; nt < 8; ++nt) {
      const float w  = acc[nt][r];
      const int   mm = mbase + nt * 16 + Lr;
      if (w > v) { v = w; bm = mm; }     // strict > keeps lowest m on ties
    }
    unsigned long long pk = ((unsigned long long)f32_key(v) << 32)
                          | (unsigned int)(~(unsigned int)bm);  // ~m: max picks lowest m
    #pragma unroll
    for (int off = 1; off < 16; off <<= 1) {
      const unsigned long long o = shflx64(pk, off);
      if (o > pk) pk = o;
    }
    if (Lr == 0) {
      const int bb = bbase + wv * 16 + half * 8 + r;
      atomicMax(&am[(size_t)bb * H_ + h], pk);
    }
  }
}

// ---------------- kernel 4: gather winning row, project, renormalize ----------------
__global__ void k4_out(const float* __restrict__ Wm, const float* __restrict__ bin_proj,
                       const unsigned long long* __restrict__ am, float* __restrict__ out) {
  const int bh = blockIdx.x;            // b*H + h
  const int b = bh >> 3, h = bh & 7;
  const int d = threadIdx.x;            // 0..63
  const unsigned long long p = am[bh];
  const int m = (int)(~(unsigned int)p) & (M_ - 1);
  const float* wrow = Wm + ((size_t)h * M_ + m) * BIN_;
  const float* bp   = bin_proj + (size_t)h * BIN_ * HD_ + d;
  float acc = 0.f;
  for (int bin = 0; bin < BIN_; ++bin)
    acc = fmaf(wrow[bin], bp[(size_t)bin * HD_], acc);
  __shared__ float sh[HD_];
  sh[d] = acc * acc;
  __syncthreads();
  float s = 0.f;
  #pragma unroll
  for (int i = 0; i < HD_; ++i) s += sh[i];
  out[(size_t)b * (H_ * HD_) + h * HD_ + d] = acc * (8.0f / sqrtf(s));  // sqrt(HD)=8
}

// ---------------- launcher ----------------
extern "C" void kernel_launch(void* const* d_in, const int* in_sizes, int n_in,
                              void* d_out, int out_size, void* d_ws, size_t ws_size,
                              hipStream_t stream) {
  const float* x        = (const float*)d_in[0];
  const float* Wq       = (const float*)d_in[1];
  const float* bq       = (const float*)d_in[2];
  const float* bin_proj = (const float*)d_in[3];
  const float* Wm       = (const float*)d_in[4];
  float* out = (float*)d_out;

  // workspace layout: [argmax u64: 64KB][q_norm f32: 2MB][mask f32: 67MB]
  char* ws = (char*)d_ws;
  unsigned long long* am = (unsigned long long*)ws;
  float* qn   = (float*)(ws + 65536);
  float* mask = (float*)(ws + 65536 + (size_t)B_ * H_ * HD_ * sizeof(float));

  hipMemsetAsync(am, 0, (size_t)B_ * H_ * sizeof(unsigned long long), stream);

  k1_qnorm<<<dim3(B_ * H_), dim3(HD_), 0, stream>>>(x, Wq, bq, qn);
  k2_topk <<<dim3(H_, B_ / 4), dim3(256), 0, stream>>>(bin_proj, qn, mask);
  k3_attn <<<dim3(B_ / 128, M_ / 128, H_), dim3(256), 0, stream>>>(Wm, mask, am);
  k4_out  <<<dim3(B_ * H_), dim3(HD_), 0, stream>>>(Wm, bin_proj, am, out);
}